// PhotonicAGPTransformer_36541581754383
// MI455X (gfx1250) — compile-verified
//
#include <hip/hip_runtime.h>
#include <math.h>

// ---------------- problem constants ----------------
static constexpr int D_FEAT = 8192;
static constexpr int T_RES  = 2048;
static constexpr int P_PAR  = 16;
static constexpr int ORDER  = 16;
static constexpr float DTAU = 0.08f;
static constexpr float REG  = 0.0001f;
static constexpr float EPSF = 1e-15f;

static constexpr int NBLK = D_FEAT / 256;        // 32 blocks over the feature dim
static constexpr int PROJ_BLOCKS = 16;           // split-K blocks for WMMA projection
static constexpr int PROJ_KCHUNK = D_FEAT / PROJ_BLOCKS; // 512

// ---------------- workspace layout (floats) ----------------
static constexpr size_t OFF_T     = 0;                      // 2048   : t = R v
static constexpr size_t OFF_W     = 2048;                   // 8192   : work vector w
static constexpr size_t OFF_Q     = 10240;                  // 17*8192: Lanczos basis
static constexpr size_t OFF_ALPHA = OFF_Q + 17 * 8192;      // 16
static constexpr size_t OFF_BETA  = OFF_ALPHA + 16;         // 16
static constexpr size_t OFF_S     = OFF_BETA + 16;          // 16 scalars (s[0]=f.f, s[2]=normF^2)
static constexpr size_t OFF_G     = OFF_S + 16;             // 256 : G = D Qb^T
static constexpr size_t OFF_DN    = OFF_G + 256;            // 16  : row norms of D
static constexpr size_t OFF_COEF  = OFF_DN + 16;            // 16  : exp(-T dtau) coeffs
static constexpr size_t OFF_APART = OFF_COEF + 16;          // 32  : block partials of q.w
static constexpr size_t OFF_NPART = OFF_APART + 32;         // 32  : block partials of ||F||^2
static constexpr size_t OFF_B2    = OFF_NPART + 32;         // 32  : block partials of ||w||^2
static constexpr size_t OFF_CPART = OFF_B2 + 32;            // 256 : split-K partials of c
static constexpr size_t WS_TOTAL  = OFF_CPART + 256;

typedef float v2f __attribute__((ext_vector_type(2)));
typedef float v8f __attribute__((ext_vector_type(8)));

// ---------------- utility ----------------
__global__ void zero_kernel(float* __restrict__ p, int n) {
  int i = blockIdx.x * blockDim.x + threadIdx.x;
  if (i < n) p[i] = 0.f;
}

// block-wide dot product of two length-8192 vectors -> *out (one block)
__global__ void dot_kernel(const float* __restrict__ x, const float* __restrict__ y,
                           float* __restrict__ out) {
  __shared__ float red[256];
  float acc = 0.f;
  for (int k = threadIdx.x; k < D_FEAT; k += 256) acc += x[k] * y[k];
  red[threadIdx.x] = acc;
  __syncthreads();
  for (int s = 128; s > 0; s >>= 1) {
    if ((int)threadIdx.x < s) red[threadIdx.x] += red[threadIdx.x + s];
    __syncthreads();
  }
  if (threadIdx.x == 0) *out = red[0];
}

// Dn[i] = sum_d D[i][d]^2     (16 blocks)
__global__ void dnorm_kernel(const float* __restrict__ Dm, float* __restrict__ Dn) {
  __shared__ float red[256];
  const float* row = Dm + (size_t)blockIdx.x * D_FEAT;
  float acc = 0.f;
  for (int k = threadIdx.x; k < D_FEAT; k += 256) { float v = row[k]; acc += v * v; }
  red[threadIdx.x] = acc;
  __syncthreads();
  for (int s = 128; s > 0; s >>= 1) {
    if ((int)threadIdx.x < s) red[threadIdx.x] += red[threadIdx.x + s];
    __syncthreads();
  }
  if (threadIdx.x == 0) Dn[blockIdx.x] = red[0];
}

// ---------------- matvecs (L2-bandwidth bound, fp32, coalesced) ----------------
// t[r] = R[r,:] . v     one wave32 per row, float4 streaming loads
__global__ void rv_matvec(const float* __restrict__ R, const float* __restrict__ v,
                          float* __restrict__ t) {
  int wave = (int)((blockIdx.x * blockDim.x + threadIdx.x) >> 5); // 0..2047
  int lane = threadIdx.x & 31;
  const float* row = R + (size_t)wave * D_FEAT;
  float acc = 0.f;
  for (int k = lane * 4; k < D_FEAT; k += 32 * 4) {
    float4 a = *(const float4*)(row + k);
    float4 b = *(const float4*)(v + k);
    acc += a.x * b.x + a.y * b.y + a.z * b.z + a.w * b.w;
  }
  for (int off = 16; off > 0; off >>= 1) acc += __shfl_down(acc, off, 32);
  if (lane == 0) t[wave] = acc;
}

// w[d] = -sum_r R[r,d] * t[r]  AND  apart[blk] = block-partial of vq . w
// (fused so the alpha/E reduction costs no extra pass over w)
__global__ void rt_matvec_neg(const float* __restrict__ R, const float* __restrict__ t,
                              const float* __restrict__ vq, float* __restrict__ w,
                              float* __restrict__ apart) {
  __shared__ float red[256];
  int d = blockIdx.x * blockDim.x + threadIdx.x; // 0..8191
  const float* p = R + d;
  float acc = 0.f;
  for (int i = 0; i < T_RES; i += 8) {
    __builtin_prefetch(p + (size_t)((i + 32) & (T_RES - 1)) * D_FEAT, 0, 0);
#pragma unroll
    for (int u = 0; u < 8; ++u) acc += p[(size_t)(i + u) * D_FEAT] * t[i + u];
  }
  float wv = -acc;
  w[d] = wv;
  red[threadIdx.x] = vq[d] * wv;
  __syncthreads();
  for (int s = 128; s > 0; s >>= 1) {
    if ((int)threadIdx.x < s) red[threadIdx.x] += red[threadIdx.x + s];
    __syncthreads();
  }
  if (threadIdx.x == 0) apart[blockIdx.x] = red[0];
}

// ---------------- Lanczos elementwise stages ----------------
// F = E*f - Hf (in place on w holding Hf); E = sum(apart)/(s[0]+eps); npart = ||F||^2 partials
__global__ void form_F(float* __restrict__ w, const float* __restrict__ f,
                       const float* __restrict__ s, const float* __restrict__ apart,
                       float* __restrict__ npart) {
  __shared__ float red[256];
  float fh = 0.f;
#pragma unroll
  for (int b = 0; b < NBLK; ++b) fh += apart[b];
  float E = fh / (s[0] + EPSF);
  int d = blockIdx.x * blockDim.x + threadIdx.x;
  float v = E * f[d] - w[d];
  w[d] = v;
  red[threadIdx.x] = v * v;
  __syncthreads();
  for (int st = 128; st > 0; st >>= 1) {
    if ((int)threadIdx.x < st) red[threadIdx.x] += red[threadIdx.x + st];
    __syncthreads();
  }
  if (threadIdx.x == 0) npart[blockIdx.x] = red[0];
}

// Q[0] = w / normF ; s2 = normF^2 (saved for the final coefficient scaling)
__global__ void set_q0(float* __restrict__ Q, const float* __restrict__ w,
                       const float* __restrict__ npart, float* __restrict__ s2) {
  float n2 = 0.f;
#pragma unroll
  for (int b = 0; b < NBLK; ++b) n2 += npart[b];
  int d = blockIdx.x * blockDim.x + threadIdx.x;
  if (d == 0) *s2 = n2;
  Q[d] = w[d] * (1.f / sqrtf(n2));
}

// alpha[j] = sum(apart); w -= alpha[j]*Q[j] + beta[j-1]*Q[j-1]
__global__ void lanczos_update(float* __restrict__ w, const float* __restrict__ Q,
                               const float* __restrict__ apart, float* __restrict__ alpha,
                               const float* __restrict__ beta, int j) {
  float a = 0.f;
#pragma unroll
  for (int b = 0; b < NBLK; ++b) a += apart[b];
  int d = blockIdx.x * blockDim.x + threadIdx.x;
  if (d == 0) alpha[j] = a;
  float bp = (j > 0) ? beta[j - 1] : 0.f;
  int jm = (j > 0) ? (j - 1) : 0;
  w[d] = w[d] - a * Q[(size_t)j * D_FEAT + d] - bp * Q[(size_t)jm * D_FEAT + d];
}

// ---- re-orth projection c = Qb . w on the WMMA pipe ----
// Split-K: block b covers K in [b*512,(b+1)*512); wave v covers 64 K (16 WMMAs).
// A (16x4) = Qb tile (row-per-lane, K pairs across 2 VGPRs, lanes 16-31 -> K+2).
// B (4x16) = w broadcast to every column  =>  C[m][n] = partial c[m] for all n.
// Extract column N=0 (lane 0 -> M=0..7, lane 16 -> M=8..15), reduce waves via LDS.
__global__ void proj_wmma(const float* __restrict__ Q, const float* __restrict__ w,
                          float* __restrict__ cpart) {
  __shared__ float part[8][16];
  int lane = threadIdx.x & 31;
  int wave = threadIdx.x >> 5; // 0..7
  int mn = lane & 15;
  int koff = (lane < 16) ? 0 : 2;
  const float* arow = Q + (size_t)mn * D_FEAT + koff;
  const float* brow = w + koff;
  v8f acc = {};
  int kb = blockIdx.x * PROJ_KCHUNK + wave * (PROJ_KCHUNK / 8);
#pragma unroll 4
  for (int it = 0; it < PROJ_KCHUNK / 8 / 4; ++it, kb += 4) {
    v2f a = *(const v2f*)(arow + kb);
    v2f b = *(const v2f*)(brow + kb);
    acc = __builtin_amdgcn_wmma_f32_16x16x4_f32(false, a, false, b, (short)0, acc,
                                                false, false);
  }
  if (lane == 0) {
#pragma unroll
    for (int r = 0; r < 8; ++r) part[wave][r] = acc[r];
  }
  if (lane == 16) {
#pragma unroll
    for (int r = 0; r < 8; ++r) part[wave][r + 8] = acc[r];
  }
  __syncthreads();
  if (threadIdx.x < 16) {
    float s = 0.f;
#pragma unroll
    for (int wv = 0; wv < 8; ++wv) s += part[wv][threadIdx.x];
    cpart[blockIdx.x * 16 + threadIdx.x] = s;
  }
}

// w -= sum_k c[k]*Q[k]  (c = deterministic sum of split-K partials); b2part = ||w||^2 partials
__global__ void subtract_proj(float* __restrict__ w, const float* __restrict__ Q,
                              const float* __restrict__ cpart, float* __restrict__ b2part) {
  __shared__ float c[16];
  __shared__ float red[256];
  if (threadIdx.x < 16) {
    float s = 0.f;
#pragma unroll
    for (int b = 0; b < PROJ_BLOCKS; ++b) s += cpart[b * 16 + threadIdx.x];
    c[threadIdx.x] = s;
  }
  __syncthreads();
  int d = blockIdx.x * blockDim.x + threadIdx.x;
  float acc = w[d];
#pragma unroll
  for (int k = 0; k < 16; ++k) acc -= c[k] * Q[(size_t)k * D_FEAT + d];
  w[d] = acc;
  red[threadIdx.x] = acc * acc;
  __syncthreads();
  for (int s = 128; s > 0; s >>= 1) {
    if ((int)threadIdx.x < s) red[threadIdx.x] += red[threadIdx.x + s];
    __syncthreads();
  }
  if (threadIdx.x == 0) b2part[blockIdx.x] = red[0];
}

// beta[j] = sqrt(sum b2part); Q[j+1] = w / max(beta[j], 1e-30)
__global__ void set_next_q(float* __restrict__ Q, const float* __restrict__ w,
                           float* __restrict__ beta, const float* __restrict__ b2part, int j) {
  float b2 = 0.f;
#pragma unroll
  for (int b = 0; b < NBLK; ++b) b2 += b2part[b];
  float bv = sqrtf(b2);
  int d = blockIdx.x * blockDim.x + threadIdx.x;
  if (d == 0) beta[j] = bv;
  float inv = 1.f / fmaxf(bv, 1e-30f);
  Q[(size_t)(j + 1) * D_FEAT + d] = w[d] * inv;
}

// ---------------- tiny 16x16 tridiagonal eigensolver + exp coefficients ----------------
__global__ void eigh_coeffs(const float* __restrict__ alpha, const float* __restrict__ beta,
                            const float* __restrict__ s, float* __restrict__ coeffs) {
  if (threadIdx.x != 0 || blockIdx.x != 0) return;
  const int n = ORDER;
  float d[ORDER], e[ORDER], V[ORDER][ORDER];
  for (int i = 0; i < n; ++i) {
    d[i] = alpha[i];
    e[i] = (i < n - 1) ? beta[i] : 0.f;
    for (int k = 0; k < n; ++k) V[i][k] = (i == k) ? 1.f : 0.f;
  }
  for (int l = 0; l < n; ++l) {
    int iter = 0;
    for (;;) {
      int m;
      for (m = l; m < n - 1; ++m) {
        float dd = fabsf(d[m]) + fabsf(d[m + 1]);
        if (fabsf(e[m]) <= 1.19209290e-07f * dd) break;
      }
      if (m == l || ++iter > 60) break;
      float g = (d[l + 1] - d[l]) / (2.f * e[l]);
      float r = sqrtf(g * g + 1.f);
      g = d[m] - d[l] + e[l] / (g + copysignf(r, g));
      float sn = 1.f, cs = 1.f, p = 0.f;
      for (int i = m - 1; i >= l; --i) {
        float fv = sn * e[i], b = cs * e[i];
        r = sqrtf(fv * fv + g * g);
        e[i + 1] = r;
        if (r == 0.f) { d[i + 1] -= p; e[m] = 0.f; break; }
        sn = fv / r; cs = g / r;
        g = d[i + 1] - p;
        r = (d[i] - g) * sn + 2.f * cs * b;
        p = sn * r;
        d[i + 1] = g + p;
        g = cs * r - b;
        for (int k = 0; k < n; ++k) {
          fv = V[k][i + 1];
          V[k][i + 1] = sn * V[k][i] + cs * fv;
          V[k][i]     = cs * V[k][i] - sn * fv;
        }
      }
      d[l] -= p; e[l] = g; e[m] = 0.f;
    }
  }
  float normF = sqrtf(s[2]);
  for (int l = 0; l < n; ++l) {
    float acc = 0.f;
    for (int m = 0; m < n; ++m) acc += V[l][m] * expf(-d[m] * DTAU) * V[0][m];
    coeffs[l] = normF * acc;
  }
}

// ---------------- G = D * Qb^T via V_WMMA_F32_16X16X4_F32 (exact fp32, K=8192) -------------
__global__ void gram_wmma(const float* __restrict__ Dm, const float* __restrict__ Q,
                          float* __restrict__ G) {
  __shared__ float red[8][32][8];
  int lane = threadIdx.x & 31;
  int wave = threadIdx.x >> 5; // 0..7
  int mn = lane & 15;
  int koff = (lane < 16) ? 0 : 2;
  const float* arow = Dm + (size_t)mn * D_FEAT + koff; // A row = D[mn, :]
  const float* brow = Q  + (size_t)mn * D_FEAT + koff; // B col = Qb[mn, :]  (B = Qb^T)
  v8f acc = {};
  int kb = wave * (D_FEAT / 8);
#pragma unroll 4
  for (int it = 0; it < D_FEAT / 8 / 4; ++it, kb += 4) {
    v2f a = *(const v2f*)(arow + kb);
    v2f b = *(const v2f*)(brow + kb);
    acc = __builtin_amdgcn_wmma_f32_16x16x4_f32(false, a, false, b, (short)0, acc,
                                                false, false);
  }
#pragma unroll
  for (int r = 0; r < 8; ++r) red[wave][lane][r] = acc[r];
  __syncthreads();
  int r  = threadIdx.x & 7;
  int l2 = threadIdx.x >> 3; // 0..31
  float sum = 0.f;
#pragma unroll
  for (int wv = 0; wv < 8; ++wv) sum += red[wv][l2][r];
  int m = (l2 < 16) ? r : (r + 8);
  int n = l2 & 15;
  G[m * 16 + n] = sum;
}

// dtheta[i] = (sum_j coeffs[j] * G[i][j]) / (Dn[i] + REG)
__global__ void final_dtheta(const float* __restrict__ G, const float* __restrict__ coeffs,
                             const float* __restrict__ Dn, float* __restrict__ out) {
  int i = threadIdx.x;
  if (i >= P_PAR) return;
  float acc = 0.f;
#pragma unroll
  for (int j = 0; j < 16; ++j) acc += coeffs[j] * G[i * 16 + j];
  out[i] = acc / (Dn[i] + REG);
}

// ---------------- launcher ----------------
extern "C" void kernel_launch(void* const* d_in, const int* in_sizes, int n_in,
                              void* d_out, int out_size, void* d_ws, size_t ws_size,
                              hipStream_t stream) {
  (void)in_sizes; (void)n_in; (void)out_size; (void)ws_size;
  const float* f  = (const float*)d_in[0];
  const float* R  = (const float*)d_in[1];
  const float* Dm = (const float*)d_in[2];
  float* out = (float*)d_out;

  float* ws    = (float*)d_ws;
  float* tvec  = ws + OFF_T;
  float* w     = ws + OFF_W;
  float* Q     = ws + OFF_Q;
  float* alpha = ws + OFF_ALPHA;
  float* beta  = ws + OFF_BETA;
  float* s     = ws + OFF_S;
  float* G     = ws + OFF_G;
  float* Dn    = ws + OFF_DN;
  float* coef  = ws + OFF_COEF;
  float* apart = ws + OFF_APART;
  float* npart = ws + OFF_NPART;
  float* b2p   = ws + OFF_B2;
  float* cpart = ws + OFF_CPART;

  // zero Q + all small state (deterministic per call; ws is poisoned otherwise)
  int nz = (int)(WS_TOTAL - OFF_Q);
  zero_kernel<<<(nz + 255) / 256, 256, 0, stream>>>(Q, nz);

  dnorm_kernel<<<P_PAR, 256, 0, stream>>>(Dm, Dn);
  dot_kernel<<<1, 256, 0, stream>>>(f, f, s + 0);

  // Hf = -R^T (R f);  E = (f.Hf)/(f.f+eps);  F = E f - Hf;  Q0 = F/||F||
  rv_matvec<<<T_RES / 8, 256, 0, stream>>>(R, f, tvec);
  rt_matvec_neg<<<NBLK, 256, 0, stream>>>(R, tvec, f, w, apart); // w=Hf, apart: f.Hf
  form_F<<<NBLK, 256, 0, stream>>>(w, f, s, apart, npart);
  set_q0<<<NBLK, 256, 0, stream>>>(Q, w, npart, s + 2);

  // 16 Lanczos steps with full WMMA re-orthogonalization
  for (int j = 0; j < ORDER; ++j) {
    const float* qj = Q + (size_t)j * D_FEAT;
    rv_matvec<<<T_RES / 8, 256, 0, stream>>>(R, qj, tvec);
    rt_matvec_neg<<<NBLK, 256, 0, stream>>>(R, tvec, qj, w, apart); // w=Hq, apart: q.w
    lanczos_update<<<NBLK, 256, 0, stream>>>(w, Q, apart, alpha, beta, j);
    proj_wmma<<<PROJ_BLOCKS, 256, 0, stream>>>(Q, w, cpart);
    subtract_proj<<<NBLK, 256, 0, stream>>>(w, Q, cpart, b2p);
    set_next_q<<<NBLK, 256, 0, stream>>>(Q, w, beta, b2p, j);
  }

  // eigh(T) + exp weighting -> coeffs ; G = D Qb^T (WMMA) ; dtheta
  eigh_coeffs<<<1, 32, 0, stream>>>(alpha, beta, s, coef);
  gram_wmma<<<1, 256, 0, stream>>>(Dm, Q, G);
  final_dtheta<<<1, 32, 0, stream>>>(G, coef, Dn, out);
}